// FNOnd_37220186587560
// MI455X (gfx1250) — compile-verified
//
#include <hip/hip_runtime.h>
#include <math.h>

// ---------------- FNO on gfx1250 with f32 WMMA (16x16x4) ----------------
// Shapes (compile-time): B=4, C=32, H=W=256, M1=M2=16, out_c=4, n_blocks=4.
// Activations are kept channels-last: [b, y, x, c] with c=32 fastest.

typedef float v2f __attribute__((ext_vector_type(2)));
typedef float v8f __attribute__((ext_vector_type(8)));

#define B_  4
#define C_  32
#define H_  256
#define W_  256
#define M_  16
#define PIX (H_ * W_)

__device__ __forceinline__ v8f wmma4(v2f a, v2f b, v8f c) {
  // D = A(16x4, f32) * B(4x16, f32) + C(16x16, f32)
  return __builtin_amdgcn_wmma_f32_16x16x4_f32(false, a, false, b, (short)0, c,
                                               false, false);
}

// A-tile 16x4: A[r][k] = p[r*lda + k]. lanes 0-15: K={0,1}; lanes 16-31: K={2,3}
__device__ __forceinline__ v2f ldA(const float* p, int lda, int lane) {
  int r = lane & 15, kk = (lane >> 4) << 1;
  v2f v; v.x = p[r * lda + kk]; v.y = p[r * lda + kk + 1]; return v;
}
// A-tile, transposed source: A[r][k] = p[k*lda + r]
__device__ __forceinline__ v2f ldAt(const float* p, int lda, int lane) {
  int r = lane & 15, kk = (lane >> 4) << 1;
  v2f v; v.x = p[kk * lda + r]; v.y = p[(kk + 1) * lda + r]; return v;
}
// B-tile 4x16: B[k][n] = p[k*ldb + n]
__device__ __forceinline__ v2f ldB(const float* p, int ldb, int lane) {
  int n = lane & 15, kk = (lane >> 4) << 1;
  v2f v; v.x = p[kk * ldb + n]; v.y = p[(kk + 1) * ldb + n]; return v;
}
// B-tile, transposed source: B[k][n] = p[n*ldb + k]
__device__ __forceinline__ v2f ldBt(const float* p, int ldb, int lane) {
  int n = lane & 15, kk = (lane >> 4) << 1;
  v2f v; v.x = p[n * ldb + kk]; v.y = p[n * ldb + kk + 1]; return v;
}
// D-tile 16x16 store: VGPR r -> row (r + 8*(lane>>4)), col (lane&15)
__device__ __forceinline__ void stD(float* p, int ldd, int lane, v8f d, float s) {
  int n = lane & 15, mh = (lane >> 4) << 3;
#pragma unroll
  for (int r = 0; r < 8; ++r) p[(r + mh) * ldd + n] = d[r] * s;
}

__device__ __forceinline__ v8f vzero() {
  v8f z = {0.f, 0.f, 0.f, 0.f, 0.f, 0.f, 0.f, 0.f};
  return z;
}

// ---------------- basis tables: cos/sin(2*pi*k*x/256) ----------------
__global__ void k_tabs(float* tabC, float* tabS, float* tabC2, float* tabS2) {
  int i = blockIdx.x * blockDim.x + threadIdx.x;  // 16*256
  if (i >= M_ * W_) return;
  int k = i >> 8, x = i & 255;
  int m = (k * x) & 255;                       // periodic reduction, exact
  float ang = 0.0245436926f * (float)m;        // 2*pi/256
  float c = cosf(ang), s = sinf(ang);
  tabC[i] = c; tabS[i] = s;
  float w2 = (k == 0) ? 1.0f : 2.0f;           // Hermitian doubling, irfft axis
  tabC2[i] = w2 * c * 0.0625f;                 // w2 * cos / 16
  tabS2[i] = w2 * s * 0.0625f;
}

// ---------------- lift: [b,ci=3,y,x] -> channels-last x0 [b,y,x,c=32] ------
__global__ void k_lift(const float* __restrict__ x, const float* __restrict__ lw,
                       const float* __restrict__ lb, float* __restrict__ x0) {
  int p = blockIdx.x * blockDim.x + threadIdx.x;  // B*PIX
  if (p >= B_ * PIX) return;
  int b = p >> 16, pix = p & (PIX - 1);
  float xi0 = x[(b * 3 + 0) * PIX + pix];
  float xi1 = x[(b * 3 + 1) * PIX + pix];
  float xi2 = x[(b * 3 + 2) * PIX + pix];
  float* o = x0 + (size_t)p * C_;
#pragma unroll
  for (int c = 0; c < C_; ++c)
    o[c] = lb[c] + lw[c * 3 + 0] * xi0 + lw[c * 3 + 1] * xi1 + lw[c * 3 + 2] * xi2;
}

// ---------------- forward DFT along x ----------------
// Tr[b,y,k2,c] = (1/16) sum_x cos(k2,x)*xin[b,y,x,c]; Ti = -(1/16) sum_x sin*xin
// one wave per (b,y,c-half): M=16 cos + 16 sin, K=256, N=16
__global__ void k_fwdX(const float* __restrict__ xin, const float* __restrict__ tabC,
                       const float* __restrict__ tabS, float* __restrict__ Tr,
                       float* __restrict__ Ti) {
  int wid = blockIdx.x * (blockDim.x >> 5) + (threadIdx.x >> 5);  // 2048 waves
  int lane = threadIdx.x & 31;
  int ch = wid & 1;            // channel half
  int byi = wid >> 1;
  int b = byi >> 8, y = byi & 255;
  const float* row = xin + ((size_t)(b * H_ + y)) * W_ * C_ + 16 * ch;  // [x][c]
  v8f aC = vzero(), aS = vzero();
  for (int k0 = 0; k0 < W_; k0 += 4) {
    v2f aCr = ldA(tabC + k0, W_, lane);   // A[k2][x]
    v2f aSr = ldA(tabS + k0, W_, lane);
    v2f bh = ldB(row + k0 * C_, C_, lane);
    aC = wmma4(aCr, bh, aC);
    aS = wmma4(aSr, bh, aS);
  }
  float* tr = Tr + ((size_t)(b * H_ + y)) * M_ * C_ + 16 * ch;  // [k2][c]
  float* ti = Ti + ((size_t)(b * H_ + y)) * M_ * C_ + 16 * ch;
  stD(tr, C_, lane, aC, 0.0625f);
  stD(ti, C_, lane, aS, -0.0625f);
}

// ---------------- forward DFT along y (K split in 4 partials) ----------------
// Frp[kc,b,k1,(k2,c)] = sum_{y in chunk kc} (cos*Tr + sin*Ti)   (scaled 1/16)
// Fip likewise with (cos*Ti - sin*Tr). One wave per (b, ntile, kc): 512 waves.
__global__ void k_fwdY(const float* __restrict__ Tr, const float* __restrict__ Ti,
                       const float* __restrict__ tabC, const float* __restrict__ tabS,
                       float* __restrict__ Frp, float* __restrict__ Fip) {
  int wid = blockIdx.x * (blockDim.x >> 5) + (threadIdx.x >> 5);  // 512 waves
  int lane = threadIdx.x & 31;
  int b = wid >> 7;
  int r2 = wid & 127;
  int col0 = (r2 >> 2) * 16;   // 32 n-tiles
  int kc = r2 & 3;             // K chunk: y in [kc*64, kc*64+64)
  const float* trb = Tr + (size_t)b * H_ * (M_ * C_);
  const float* tib = Ti + (size_t)b * H_ * (M_ * C_);
  v8f aCT = vzero(), aST = vzero(), aCTi = vzero(), aSTr = vzero();
  for (int k0 = kc * 64; k0 < kc * 64 + 64; k0 += 4) {
    v2f aC = ldA(tabC + k0, W_, lane);  // A[k1][y]
    v2f aS = ldA(tabS + k0, W_, lane);
    v2f bR = ldB(trb + k0 * 512 + col0, 512, lane);
    v2f bI = ldB(tib + k0 * 512 + col0, 512, lane);
    aCT = wmma4(aC, bR, aCT);
    aST = wmma4(aS, bI, aST);
    aCTi = wmma4(aC, bI, aCTi);
    aSTr = wmma4(aS, bR, aSTr);
  }
  float* fr = Frp + (size_t)kc * (B_ * M_ * 512) + (size_t)b * (M_ * 512) + col0;
  float* fi = Fip + (size_t)kc * (B_ * M_ * 512) + (size_t)b * (M_ * 512) + col0;
  int n = lane & 15, mh = (lane >> 4) << 3;
#pragma unroll
  for (int r = 0; r < 8; ++r) {
    fr[(r + mh) * 512 + n] = (aCT[r] + aST[r]) * 0.0625f;
    fi[(r + mh) * 512 + n] = (aCTi[r] - aSTr[r]) * 0.0625f;
  }
}

// ---------------- spectral 32x32 complex channel mix (per mode) ----------
// Sums the 4 fwdY partials, stages F-slice in LDS, coalesced weight reads.
// G[b,k1,k2,o] = sum_i F[b,k1,k2,i] * (wr+j*wi)[i,o,k1,k2]
// one block of 512 threads per (b,k1); thread = (o = tid>>4, k2 = tid&15)
__global__ void k_mix(const float* __restrict__ Frp, const float* __restrict__ Fip,
                      const float* __restrict__ wr, const float* __restrict__ wi,
                      float* __restrict__ Gr, float* __restrict__ Gi) {
  __shared__ float sFr[512];
  __shared__ float sFi[512];
  int b = blockIdx.x >> 4, k1 = blockIdx.x & 15;  // grid 64
  int tid = threadIdx.x;                           // 512
  size_t base = ((size_t)b * M_ + k1) * 512;
  float fr = 0.f, fi = 0.f;
#pragma unroll
  for (int kc = 0; kc < 4; ++kc) {
    fr += Frp[(size_t)kc * (B_ * M_ * 512) + base + tid];
    fi += Fip[(size_t)kc * (B_ * M_ * 512) + base + tid];
  }
  sFr[tid] = fr; sFi[tid] = fi;
  __syncthreads();
  int o = tid >> 4, k2 = tid & 15;
  const float* wrk = wr + k1 * 16 + k2;  // + (i*32+o)*256
  const float* wik = wi + k1 * 16 + k2;
  float gr = 0.f, gi = 0.f;
#pragma unroll 4
  for (int i = 0; i < C_; ++i) {
    float xr = sFr[k2 * 32 + i], xi = sFi[k2 * 32 + i];
    float wrv = wrk[(i * 32 + o) * 256];
    float wiv = wik[(i * 32 + o) * 256];
    gr += xr * wrv - xi * wiv;
    gi += xr * wiv + xi * wrv;
  }
  size_t gidx = base + k2 * 32 + o;
  Gr[gidx] = gr; Gi[gidx] = gi;
}

// ---------------- inverse DFT along y ----------------
// P[b,y,(k2,c)] = (1/16) sum_k1 (cos*Gr - sin*Gi); Q = (1/16) sum_k1 (sin*Gr + cos*Gi)
// one wave per (b, ytile, ntile): M=16(y), K=16(k1), N=16 of 512
__global__ void k_invY(const float* __restrict__ Gr, const float* __restrict__ Gi,
                       const float* __restrict__ tabC, const float* __restrict__ tabS,
                       float* __restrict__ P, float* __restrict__ Q) {
  int wid = blockIdx.x * (blockDim.x >> 5) + (threadIdx.x >> 5);
  int lane = threadIdx.x & 31;
  int b = wid >> 9, rest = wid & 511;
  int y0 = (rest >> 5) * 16, col0 = (rest & 31) * 16;
  const float* grb = Gr + (size_t)b * 8192;
  const float* gib = Gi + (size_t)b * 8192;
  v8f aCG = vzero(), aSGi = vzero(), aSG = vzero(), aCGi = vzero();
#pragma unroll
  for (int k0 = 0; k0 < M_; k0 += 4) {
    v2f aC = ldAt(tabC + k0 * W_ + y0, W_, lane);  // A[y][k1] = tabC[k1*256+y]
    v2f aS = ldAt(tabS + k0 * W_ + y0, W_, lane);
    v2f bR = ldB(grb + k0 * 512 + col0, 512, lane);
    v2f bI = ldB(gib + k0 * 512 + col0, 512, lane);
    aCG = wmma4(aC, bR, aCG);
    aSGi = wmma4(aS, bI, aSGi);
    aSG = wmma4(aS, bR, aSG);
    aCGi = wmma4(aC, bI, aCGi);
  }
  float* p = P + (size_t)(b * H_ + y0) * 512 + col0;
  float* q = Q + (size_t)(b * H_ + y0) * 512 + col0;
  int n = lane & 15, mh = (lane >> 4) << 3;
#pragma unroll
  for (int r = 0; r < 8; ++r) {
    p[(r + mh) * 512 + n] = (aCG[r] - aSGi[r]) * 0.0625f;
    q[(r + mh) * 512 + n] = (aSG[r] + aCGi[r]) * 0.0625f;
  }
}

// ---------------- inverse DFT along x + bypass conv1x1 + bias + GELU -----
// out[b,y,x,c] = gelu( sum_k2 (tabC2[k2,x]*P - tabS2[k2,x]*Q)
//                      + bb[c] + sum_ci bw[c,ci]*xin[b,y,x,ci] )
// one wave per (b,y,xtile): two 16x16 D tiles (c halves)
__global__ void k_invX(const float* __restrict__ P, const float* __restrict__ Q,
                       const float* __restrict__ xin, const float* __restrict__ tabC2,
                       const float* __restrict__ tabS2, const float* __restrict__ bw,
                       const float* __restrict__ bb, float* __restrict__ xout) {
  int wid = blockIdx.x * (blockDim.x >> 5) + (threadIdx.x >> 5);
  int lane = threadIdx.x & 31;
  int by = wid >> 4;            // b*256 + y
  int x0 = (wid & 15) * 16;
  const float* Pby = P + (size_t)by * 512;
  const float* Qby = Q + (size_t)by * 512;
  const float* xrow = xin + (size_t)by * (W_ * C_);
  v8f acc0 = vzero(), acc1 = vzero();
  // spectral part: K = 16 (P with cos) + 16 (Q with -sin)
#pragma unroll
  for (int k0 = 0; k0 < M_; k0 += 4) {
    v2f aC = ldAt(tabC2 + k0 * W_ + x0, W_, lane);  // A[x][k2]
    v2f bP0 = ldB(Pby + k0 * C_, C_, lane);
    v2f bP1 = ldB(Pby + k0 * C_ + 16, C_, lane);
    acc0 = wmma4(aC, bP0, acc0);
    acc1 = wmma4(aC, bP1, acc1);
    v2f aS = ldAt(tabS2 + k0 * W_ + x0, W_, lane);
    v2f aSn; aSn.x = -aS.x; aSn.y = -aS.y;
    v2f bQ0 = ldB(Qby + k0 * C_, C_, lane);
    v2f bQ1 = ldB(Qby + k0 * C_ + 16, C_, lane);
    acc0 = wmma4(aSn, bQ0, acc0);
    acc1 = wmma4(aSn, bQ1, acc1);
  }
  // bypass part: K = 32 over ci; A[x][ci] coalesced; B[ci][c] = bw[c][ci]
#pragma unroll
  for (int k0 = 0; k0 < C_; k0 += 4) {
    v2f aX = ldA(xrow + x0 * C_ + k0, C_, lane);
    v2f bW0 = ldBt(bw + k0, C_, lane);            // c 0..15
    v2f bW1 = ldBt(bw + 16 * C_ + k0, C_, lane);  // c 16..31
    acc0 = wmma4(aX, bW0, acc0);
    acc1 = wmma4(aX, bW1, acc1);
  }
  int n = lane & 15, mh = (lane >> 4) << 3;
  float bb0 = bb[n], bb1 = bb[16 + n];
  float* out = xout + (size_t)by * (W_ * C_);
#pragma unroll
  for (int r = 0; r < 8; ++r) {
    int xr = x0 + r + mh;
    float v0 = acc0[r] + bb0;
    v0 = 0.5f * v0 * (1.0f + erff(v0 * 0.70710678f));
    out[xr * C_ + n] = v0;
    float v1 = acc1[r] + bb1;
    v1 = 0.5f * v1 * (1.0f + erff(v1 * 0.70710678f));
    out[xr * C_ + 16 + n] = v1;
  }
}

// ---------------- projection: dot(xb[b,y,x,:], proj_w) + proj_b ----------
__global__ void k_proj(const float* __restrict__ xb, const float* __restrict__ pw,
                       const float* __restrict__ pb, float* __restrict__ out,
                       int o_outer) {
  int p = blockIdx.x * blockDim.x + threadIdx.x;  // B*PIX
  if (p >= B_ * PIX) return;
  int b = p >> 16, pix = p & (PIX - 1);
  const float* v = xb + (size_t)p * C_;
  float s = pb[0];
#pragma unroll
  for (int c = 0; c < C_; ++c) s += pw[c] * v[c];
  out[((size_t)(b * 4 + o_outer)) * PIX + pix] = s;
}

extern "C" void kernel_launch(void* const* d_in, const int* in_sizes, int n_in,
                              void* d_out, int out_size, void* d_ws, size_t ws_size,
                              hipStream_t stream) {
  const float* x      = (const float*)d_in[0];  // [4,3,256,256]
  const float* lift_w = (const float*)d_in[1];  // [32,3]
  const float* lift_b = (const float*)d_in[2];  // [32]
  const float* wr     = (const float*)d_in[3];  // [4,4,32,32,16,16]
  const float* wi     = (const float*)d_in[4];
  const float* byp_w  = (const float*)d_in[5];  // [4,4,32,32]
  const float* byp_b  = (const float*)d_in[6];  // [4,4,32]
  const float* proj_w = (const float*)d_in[7];  // [1,32]
  const float* proj_b = (const float*)d_in[8];  // [1]
  float* out = (float*)d_out;

  // workspace carve-up (floats); total ~27.6M floats (~110 MB)
  float* w = (float*)d_ws;
  float* tabC  = w;                 // 16*256
  float* tabS  = tabC + 4096;
  float* tabC2 = tabS + 4096;
  float* tabS2 = tabC2 + 4096;
  float* x0    = tabS2 + 4096;      // B*PIX*C
  float* bufA  = x0 + (size_t)B_ * PIX * C_;
  float* bufB  = bufA + (size_t)B_ * PIX * C_;
  float* Tr    = bufB + (size_t)B_ * PIX * C_;   // B*H*16*32
  float* Ti    = Tr + (size_t)B_ * H_ * M_ * C_;
  float* Frp   = Ti + (size_t)B_ * H_ * M_ * C_; // 4 partials * B*16*512
  float* Fip   = Frp + (size_t)4 * B_ * M_ * 512;
  float* Gr    = Fip + (size_t)4 * B_ * M_ * 512;
  float* Gi    = Gr + (size_t)B_ * M_ * M_ * C_;
  float* Pb    = Gi + (size_t)B_ * M_ * M_ * C_; // B*H*16*32
  float* Qb    = Pb + (size_t)B_ * H_ * M_ * C_;

  k_tabs<<<16, 256, 0, stream>>>(tabC, tabS, tabC2, tabS2);
  k_lift<<<(B_ * PIX + 255) / 256, 256, 0, stream>>>(x, lift_w, lift_b, x0);

  for (int o = 0; o < 4; ++o) {
    for (int n = 0; n < 4; ++n) {
      const float* cur = (n == 0) ? x0 : ((n & 1) ? bufA : bufB);
      float* dst = (n & 1) ? bufB : bufA;
      size_t wOff = ((size_t)(o * 4 + n)) * C_ * C_ * M_ * M_;
      size_t bwOff = ((size_t)(o * 4 + n)) * C_ * C_;
      size_t bbOff = ((size_t)(o * 4 + n)) * C_;
      k_fwdX<<<256, 256, 0, stream>>>(cur, tabC, tabS, Tr, Ti);
      k_fwdY<<<64, 256, 0, stream>>>(Tr, Ti, tabC, tabS, Frp, Fip);
      k_mix<<<64, 512, 0, stream>>>(Frp, Fip, wr + wOff, wi + wOff, Gr, Gi);
      k_invY<<<256, 256, 0, stream>>>(Gr, Gi, tabC, tabS, Pb, Qb);
      k_invX<<<2048, 256, 0, stream>>>(Pb, Qb, cur, tabC2, tabS2,
                                       byp_w + bwOff, byp_b + bbOff, dst);
    }
    k_proj<<<(B_ * PIX + 255) / 256, 256, 0, stream>>>(bufB, proj_w, proj_b, out, o);
  }
}